// ODEModelLayer_28879360099043
// MI455X (gfx1250) — compile-verified
//
#include <hip/hip_runtime.h>
#include <stdint.h>

#define THREADS 256
#define NSTEPS  64

typedef float f4  __attribute__((ext_vector_type(4)));
typedef float f2  __attribute__((ext_vector_type(2)));
typedef int   v4i __attribute__((vector_size(16)));   // matches builtin's V4i

__device__ __forceinline__ f2 mk2(float a, float b) { f2 r; r.x = a; r.y = b; return r; }

// gfx1250 async global->LDS path (ASYNCcnt), with portable fallback.
#if defined(__gfx1250__) && __has_builtin(__builtin_amdgcn_global_load_async_to_lds_b128)
#define USE_ASYNC_LDS 1
#else
#define USE_ASYNC_LDS 0
#endif

__device__ __forceinline__ void cp16_g2lds(const float* __restrict__ g, float* l) {
#if USE_ASYNC_LDS
  __builtin_amdgcn_global_load_async_to_lds_b128(
      (__attribute__((address_space(1))) v4i*)(uintptr_t)g,
      (__attribute__((address_space(3))) v4i*)(uint32_t)(uintptr_t)l,
      /*offset=*/0, /*cpol=*/0);
#else
  *(f4*)l = *(const f4*)g;
#endif
}

__device__ __forceinline__ void wait_async_then_barrier() {
#if USE_ASYNC_LDS
#if __has_builtin(__builtin_amdgcn_s_wait_asynccnt)
  __builtin_amdgcn_s_wait_asynccnt(0);
#else
  asm volatile("s_wait_asynccnt 0" ::: "memory");
#endif
#endif
  __syncthreads();
}

// Dormand-Prince tableau (fp32 literals)
#define A21 0.2f
#define A31 0.075f
#define A32 0.225f
#define A41 (44.0f/45.0f)
#define A42 (-56.0f/15.0f)
#define A43 (32.0f/9.0f)
#define A51 (19372.0f/6561.0f)
#define A52 (-25360.0f/2187.0f)
#define A53 (64448.0f/6561.0f)
#define A54 (-212.0f/729.0f)
#define A61 (9017.0f/3168.0f)
#define A62 (-355.0f/33.0f)
#define A63 (46732.0f/5247.0f)
#define A64 (49.0f/176.0f)
#define A65 (-5103.0f/18656.0f)
#define B1c (35.0f/384.0f)
#define B3c (500.0f/1113.0f)
#define B4c (125.0f/192.0f)
#define B5c (-2187.0f/6784.0f)
#define B6c (11.0f/84.0f)

// Per-sample coefficients, pre-paired for packed-f32 math and pre-scaled by h
// where the coefficient touches raw state exactly once (so rhs returns h*f).
struct Coef {
  f2 p01, p23, p45, p67;     // h-scaled rates: (p0,p1),(p2,p3),(p4,p5),(p6,p7)
  f2 c45, c06, c78;          // (h*D*z4, h*D*z5), (0, h*D*z6), (h*D*z7, h*D*z8)
  f2 nD2;                    // (-h*D, -h*D)
  f2 d1D;                    // (h*d1, h*D)
  f2 p14_22, p20_21;         // unscaled (multiply already-scaled r terms)
  f2 p8_11, p9_12, p10_13;   // unscaled
  f2 p15_16, p17_18;         // h-scaled
  float p19;                 // h-scaled
};

// Pair layout:  v0=(X,Prod)  v1=(Glc,Gln)  v2=(Lac,Glu)  v3=(NH4,Osmo)
//               k0=(dX,dProd) k1=(dGlc,dGln) k2=(dLac,dGlu) k3=(dNH4,dOsmo)
__device__ __forceinline__ void rhs_h(const f2 v[4], const Coef& C, f2 k[4]) {
  const float X = v[0].x;
  const f2 Xs  = mk2(X, X);
  const f2 r01 = (C.p01 * Xs) * v[1];          // (rGlc, rGln)   (h-scaled)
  const f2 r23 = (C.p23 * Xs) * v[2];          // (rLac, rGlu)
  const f2 r45 = (C.p45 * Xs) * v[3];          // (rNH4, rOsmo)
  const f2 rkd = C.p67 * v[0];                 // (kD, rDegP)

  // (dGlc, dGln) = (c4,c5) - D*(Glc,Gln) - (rGlc,rGln)
  k[1] = (C.nD2 * v[1] + C.c45) - r01;

  // (dLac, dGlu) = (rGlc*p14, c6 + rGln*p22) - D*(Lac,Glu) - (rLac,rGlu)
  f2 m = r01 * C.p14_22 + C.c06;
  k[2] = (C.nD2 * v[2] + m) - r23;

  // (dNH4, dOsmo)
  f2 r01s = mk2(r01.y, r01.x);                 // (rGln, rGlc) via OPSEL swap
  f2 e    = r01s * C.p20_21;                   // (rGln*p20, rGlc*p21)
  const float extra5 = e.x + e.y;
  f2 a = v[1] * C.p15_16;                      // (Glc*p15, Gln*p16)
  a = v[2] * C.p17_18 + a;                     // + (Lac*p17, Glu*p18)
  const float lac2osmo = (a.x + a.y) + v[3].x * C.p19;
  f2 base = (C.nD2 * v[3] + C.c78) - r45;      // (c7-D*NH4-rNH4, c8-D*Osmo-rOsmo)
  k[3] = base + mk2(extra5, lac2osmo);

  // (dX, dProd)
  f2 t = mk2(r01.x, r01.x) * C.p8_11;          // rGlc * (p8, p11)
  t = mk2(r01.y, r01.y) * C.p9_12  + t;        // + rGln * (p9, p12)
  t = mk2(r23.y, r23.y) * C.p10_13 + t;        // + rGlu * (p10, p13)
  t = t - C.d1D * v[0];                        // - (d1*X, D*Prod)
  k[0] = t - rkd;                              // - (kD, rDegP)
}

__global__ __launch_bounds__(THREADS) void rk45_dopri_kernel(
    const float* __restrict__ g_preds,   // [B,23]
    const float* __restrict__ g_const,   // [B,9]
    const float* __restrict__ g_x0,      // [B,10]
    const float* __restrict__ g_dt,      // [B]
    float* __restrict__ g_out)           // [B,10]
{
  // Coalesced b128 staging buffers (44 KB << 320 KB/WGP).
  __shared__ f4 s4_preds[23 * THREADS / 4];  // 1472
  __shared__ f4 s4_const[ 9 * THREADS / 4];  //  576
  __shared__ f4 s4_x   [10 * THREADS / 4];   //  640
  __shared__ f4 s4_dt  [     THREADS / 4];   //   64

  const int tid = threadIdx.x;
  const long long row0 = (long long)blockIdx.x * THREADS;

  {
    const float* gp = g_preds + row0 * 23;   // block bases are all 16B-aligned
    const float* gc = g_const + row0 * 9;
    const float* gx = g_x0    + row0 * 10;
    const float* gd = g_dt    + row0;
    for (int i = tid; i < 23 * THREADS / 4; i += THREADS) cp16_g2lds(gp + 4 * i, (float*)(s4_preds + i));
    for (int i = tid; i <  9 * THREADS / 4; i += THREADS) cp16_g2lds(gc + 4 * i, (float*)(s4_const + i));
    for (int i = tid; i < 10 * THREADS / 4; i += THREADS) cp16_g2lds(gx + 4 * i, (float*)(s4_x    + i));
    for (int i = tid; i <      THREADS / 4; i += THREADS) cp16_g2lds(gd + 4 * i, (float*)(s4_dt   + i));
  }
  wait_async_then_barrier();

  const float* s_preds = (const float*)s4_preds;
  const float* s_const = (const float*)s4_const;
  float*       s_x     = (float*)s4_x;
  const float* s_dt    = (const float*)s4_dt;

  const float h = s_dt[tid] * (1.0f / (float)NSTEPS);

  // Registers (strides 23/9 are odd -> LDS conflict-free).
  float p[23];
#pragma unroll
  for (int i = 0; i < 23; ++i) p[i] = s_preds[tid * 23 + i];

  const float z0 = s_const[tid * 9 + 0];
  const float z1 = s_const[tid * 9 + 1];
  const float z3 = s_const[tid * 9 + 3];
  const float z4 = s_const[tid * 9 + 4];
  const float z5 = s_const[tid * 9 + 5];
  const float z6 = s_const[tid * 9 + 6];
  const float z7 = s_const[tid * 9 + 7];
  const float z8 = s_const[tid * 9 + 8];
  const float rz0 = 1.0f / z0;                // the ONLY division for this sample
  const float hD  = z3 * rz0 * h;
  const float hd1 = z1 * rz0 * h;

  Coef C;
  C.p01    = mk2(p[0] * h,  p[1] * h);
  C.p23    = mk2(p[2] * h,  p[3] * h);
  C.p45    = mk2(p[4] * h,  p[5] * h);
  C.p67    = mk2(p[6] * h,  p[7] * h);
  C.c45    = mk2(hD * z4,   hD * z5);
  C.c06    = mk2(0.0f,      hD * z6);
  C.c78    = mk2(hD * z7,   hD * z8);
  C.nD2    = mk2(-hD,       -hD);
  C.d1D    = mk2(hd1,       hD);
  C.p14_22 = mk2(p[14],     p[22]);
  C.p20_21 = mk2(p[20],     p[21]);
  C.p8_11  = mk2(p[8],      p[11]);
  C.p9_12  = mk2(p[9],      p[12]);
  C.p10_13 = mk2(p[10],     p[13]);
  C.p15_16 = mk2(p[15] * h, p[16] * h);
  C.p17_18 = mk2(p[17] * h, p[18] * h);
  C.p19    = p[19] * h;

  // State pairs: v0=(X,Prod) v1=(Glc,Gln) v2=(Lac,Glu) v3=(NH4,Osmo)
  f2 x[4];
  x[0] = mk2(s_x[tid * 10 + 0], s_x[tid * 10 + 6]);
  x[1] = mk2(s_x[tid * 10 + 1], s_x[tid * 10 + 2]);
  x[2] = mk2(s_x[tid * 10 + 3], s_x[tid * 10 + 4]);
  x[3] = mk2(s_x[tid * 10 + 5], s_x[tid * 10 + 7]);

#pragma unroll 1
  for (int st = 0; st < NSTEPS; ++st) {
    f2 k1[4], k2[4], k3[4], k4[4], k5[4], k6[4], xi[4];
    rhs_h(x,  C, k1);                          // k = h*f already
#pragma unroll
    for (int j = 0; j < 4; ++j) xi[j] = x[j] + A21 * k1[j];
    rhs_h(xi, C, k2);
#pragma unroll
    for (int j = 0; j < 4; ++j) xi[j] = x[j] + A31 * k1[j] + A32 * k2[j];
    rhs_h(xi, C, k3);
#pragma unroll
    for (int j = 0; j < 4; ++j) xi[j] = x[j] + A41 * k1[j] + A42 * k2[j] + A43 * k3[j];
    rhs_h(xi, C, k4);
#pragma unroll
    for (int j = 0; j < 4; ++j) xi[j] = x[j] + A51 * k1[j] + A52 * k2[j] + A53 * k3[j] + A54 * k4[j];
    rhs_h(xi, C, k5);
#pragma unroll
    for (int j = 0; j < 4; ++j) xi[j] = x[j] + A61 * k1[j] + A62 * k2[j] + A63 * k3[j] + A64 * k4[j] + A65 * k5[j];
    rhs_h(xi, C, k6);
#pragma unroll
    for (int j = 0; j < 4; ++j) x[j] = x[j] + B1c * k1[j] + B3c * k3[j] + B4c * k4[j] + B5c * k5[j] + B6c * k6[j];
  }

  // Unpack back to this thread's LDS row; dims 8,9 keep original x0 values.
  s_x[tid * 10 + 0] = x[0].x;  s_x[tid * 10 + 6] = x[0].y;
  s_x[tid * 10 + 1] = x[1].x;  s_x[tid * 10 + 2] = x[1].y;
  s_x[tid * 10 + 3] = x[2].x;  s_x[tid * 10 + 4] = x[2].y;
  s_x[tid * 10 + 5] = x[3].x;  s_x[tid * 10 + 7] = x[3].y;
  __syncthreads();

  // Coalesced b128 stores of the whole block's [256 x 10] result tile.
  f4* go = (f4*)(g_out + row0 * 10);
#pragma unroll 1
  for (int i = tid; i < 10 * THREADS / 4; i += THREADS) go[i] = s4_x[i];
}

extern "C" void kernel_launch(void* const* d_in, const int* in_sizes, int n_in,
                              void* d_out, int out_size, void* d_ws, size_t ws_size,
                              hipStream_t stream) {
  const float* preds  = (const float*)d_in[0];  // [B,23]
  const float* consts = (const float*)d_in[1];  // [B,9]
  const float* x0     = (const float*)d_in[2];  // [B,10]
  const float* dt     = (const float*)d_in[3];  // [B]
  float* out = (float*)d_out;

  const int B = in_sizes[3];          // 524288, multiple of 256
  const int blocks = B / THREADS;     // 2048 blocks of 8 wave32s
  rk45_dopri_kernel<<<blocks, THREADS, 0, stream>>>(preds, consts, x0, dt, out);
}